// SelfMaskedAttention_55757265437087
// MI455X (gfx1250) — compile-verified
//
#include <hip/hip_runtime.h>

// Problem constants (from reference)
#define B_ 2
#define S_ 1024
#define CACHE_ 512
#define T_ 1536          // CACHE + S
#define E_ 1024
#define H_ 16
#define D_ 64
#define E3_ 3072

// LDS row strides padded for 64-bank conflict-free access
#define SA_STRIDE 20     // A tiles / transpose tiles (16-wide rows)
#define SB_STRIDE 80     // B / V tiles (64-wide rows)

typedef __attribute__((ext_vector_type(2))) float v2f;
typedef __attribute__((ext_vector_type(8))) float v8f;
typedef __attribute__((address_space(3))) float lds_f;

// ---------------------------------------------------------------------------
// V_WMMA_F32_16X16X4_F32:  D(16x16 f32) = A(16x4 f32) * B(4x16 f32) + C
// A: lanes 0-15 row M=lane, (a.x,a.y)=(K0,K1); lanes16-31 (K2,K3)
// B (mirror): lanes 0-15 col N=lane, (b.x,b.y)=(K0,K1); hi half (K2,K3)
// C/D: VGPR r, lanes 0-15 -> (M=r, N=lane); lanes 16-31 -> (M=r+8)
// ---------------------------------------------------------------------------
__device__ __forceinline__ v8f wmma4(v2f a, v2f b, v8f c) {
  return __builtin_amdgcn_wmma_f32_16x16x4_f32(
      false, a, false, b, (short)0, c, false, false);
}

// LDS byte offset of a __shared__ element (addrspace(3) value)
__device__ __forceinline__ uint32_t to_lds_off(float* p) {
  return (uint32_t)(uintptr_t)(lds_f*)p;
}

// One 16-byte async DMA transfer: global -> LDS, tracked by ASYNCcnt.
__device__ __forceinline__ void async_b128(float* lds_dst, const float* g) {
  uint32_t off = to_lds_off(lds_dst);
  asm volatile("global_load_async_to_lds_b128 %0, %1, off"
               :: "v"(off), "v"(g) : "memory");
}

__device__ __forceinline__ void wait_async0() {
  asm volatile("s_wait_asynccnt 0x0" ::: "memory");
}

// ---------------------------------------------------------------------------
// Shared double-buffered GEMM block: 128 rows x 64 cols per workgroup,
// 8 waves each own a 16x64 strip.  K consumed in 16-deep async-staged tiles.
//   sh_a: 2 x (128 x SA_STRIDE) floats, sh_b: 2 x (16 x SB_STRIDE) floats
// ---------------------------------------------------------------------------
__device__ __forceinline__ void issue_gemm_stage(const float* __restrict__ A, int lda,
                                                 const float* __restrict__ Bm, int ldb,
                                                 int row0, int col0, int k0,
                                                 float* sa, float* sb) {
  const int tid = threadIdx.x;
  // A tile 128x16 = 512 b128 transfers (2 per thread)
  {
    int row = tid >> 2, seg = (tid & 3) * 4;
    async_b128(sa + row * SA_STRIDE + seg, A + (size_t)(row0 + row) * lda + k0 + seg);
    int id = tid + 256;
    row = id >> 2; seg = (id & 3) * 4;
    async_b128(sa + row * SA_STRIDE + seg, A + (size_t)(row0 + row) * lda + k0 + seg);
  }
  // B tile 16x64 = 256 b128 transfers (1 per thread)
  {
    int row = tid >> 4, seg = (tid & 15) * 4;
    async_b128(sb + row * SB_STRIDE + seg, Bm + (size_t)(k0 + row) * ldb + col0 + seg);
  }
}

__device__ __forceinline__ void gemm_block_128x64(const float* __restrict__ A, int lda,
                                                  const float* __restrict__ Bm, int ldb,
                                                  int K, int row0, int col0,
                                                  float* sh_a, float* sh_b,
                                                  int wave, int l16, int lhalf,
                                                  v8f acc[4]) {
  issue_gemm_stage(A, lda, Bm, ldb, row0, col0, 0, sh_a, sh_b);
  for (int ks = 0; ks < K; ks += 16) {
    const int buf = (ks >> 4) & 1;
    wait_async0();               // my stage's DMA done
    __syncthreads();             // everyone's DMA visible; prev buf free
    if (ks + 16 < K)
      issue_gemm_stage(A, lda, Bm, ldb, row0, col0, ks + 16,
                       sh_a + (buf ^ 1) * (128 * SA_STRIDE),
                       sh_b + (buf ^ 1) * (16 * SB_STRIDE));
    const float* sa = sh_a + buf * (128 * SA_STRIDE);
    const float* sb = sh_b + buf * (16 * SB_STRIDE);
#pragma unroll
    for (int kk = 0; kk < 16; kk += 4) {
      v2f a;
      const float* ap = sa + (wave * 16 + l16) * SA_STRIDE + kk + 2 * lhalf;
      a.x = ap[0];
      a.y = ap[1];
#pragma unroll
      for (int n = 0; n < 4; ++n) {
        v2f b;
        const float* bp = sb + (kk + 2 * lhalf) * SB_STRIDE + 16 * n + l16;
        b.x = bp[0];
        b.y = bp[SB_STRIDE];
        acc[n] = wmma4(a, b, acc[n]);
      }
    }
  }
}

// ---------------------------------------------------------------------------
// Kernel 1: qkv = x @ W_qkv + b_qkv, scattered to q_ws / k_full / v_full
//   Grid: (2048/128) x (3072/64) = 16*48 = 768 blocks of 256 threads.
// ---------------------------------------------------------------------------
__global__ void __launch_bounds__(256) qkv_kernel(const float* __restrict__ x,
                                                  const float* __restrict__ Wqkv,
                                                  const float* __restrict__ bqkv,
                                                  float* __restrict__ q_ws,
                                                  float* __restrict__ kfull,
                                                  float* __restrict__ vfull) {
  __shared__ float sh_a[2 * 128 * SA_STRIDE];
  __shared__ float sh_b[2 * 16 * SB_STRIDE];
  const int wave = threadIdx.x >> 5;
  const int lane = threadIdx.x & 31;
  const int l16 = lane & 15, lhalf = lane >> 4;
  const int nCB = E3_ / 64;                // 48
  const int row0 = (blockIdx.x / nCB) * 128;
  const int col0 = (blockIdx.x % nCB) * 64;

  v8f zero = {};
  v8f acc[4];
#pragma unroll
  for (int n = 0; n < 4; ++n) acc[n] = zero;

  gemm_block_128x64(x, E_, Wqkv, E3_, E_, row0, col0, sh_a, sh_b,
                    wave, l16, lhalf, acc);

  const int rowW = row0 + wave * 16;       // this wave's 16-row strip
  const int bIdx = rowW / S_;
  const int s0 = rowW % S_;
#pragma unroll
  for (int n = 0; n < 4; ++n) {
    const int col = col0 + 16 * n + l16;
    const float bias = bqkv[col];
#pragma unroll
    for (int r = 0; r < 8; ++r) {
      const int srow = s0 + r + 8 * lhalf;
      const float val = acc[n][r] + bias;
      if (col < E_) {                      // Q -> (B,H,S,D) workspace
        const int h = col / D_, d = col % D_;
        q_ws[(((size_t)(bIdx * H_ + h) * S_) + srow) * D_ + d] = val;
      } else if (col < 2 * E_) {           // K -> k_full (B,H,D,T)
        const int cc = col - E_;
        const int h = cc / D_, d = cc % D_;
        kfull[(((size_t)(bIdx * H_ + h) * D_) + d) * T_ + (CACHE_ + srow)] = val;
      } else {                             // V -> v_full (B,H,T,D)
        const int cc = col - 2 * E_;
        const int h = cc / D_, d = cc % D_;
        vfull[(((size_t)(bIdx * H_ + h) * T_) + (CACHE_ + srow)) * D_ + d] = val;
      }
    }
  }
}

// ---------------------------------------------------------------------------
// Kernel 2: cached K/V concat copies (vectorized float4)
// ---------------------------------------------------------------------------
__global__ void __launch_bounds__(256) copy_cache_kernel(const float4* __restrict__ ck,
                                                         const float4* __restrict__ cv,
                                                         float* __restrict__ kfull,
                                                         float* __restrict__ vfull) {
  const int nk4 = (B_ * H_ * D_ * CACHE_) / 4;   // 262144
  const int nv4 = (B_ * H_ * CACHE_ * D_) / 4;   // 262144
  const int idx = blockIdx.x * blockDim.x + threadIdx.x;
  if (idx < nk4) {
    const int c4 = idx % (CACHE_ / 4);
    const int bhd = idx / (CACHE_ / 4);
    *(float4*)(kfull + (size_t)bhd * T_ + c4 * 4) = ck[idx];
  } else if (idx < nk4 + nv4) {
    const int j = idx - nk4;
    const int bh = j / ((CACHE_ * D_) / 4);
    const int rem = j % ((CACHE_ * D_) / 4);
    *(float4*)(vfull + (size_t)bh * (T_ * D_) + rem * 4) = cv[j];
  }
}

// ---------------------------------------------------------------------------
// Kernel 3: attention.  Block = one (b,h) x 128 query rows (8 waves x 16).
//   K/V tiles async-staged to LDS, double-buffered, shared by all waves.
//   Q fragments preloaded to registers and reused across all 96 T-tiles.
// Grid: B*H*(S/128) = 256 blocks of 256 threads.
// ---------------------------------------------------------------------------
__device__ __forceinline__ void issue_attn_stage(const float* __restrict__ K,
                                                 const float* __restrict__ V,
                                                 int t0, float* shk, float* shv) {
  const int tid = threadIdx.x;
  // K tile: 64 rows(d) x 16 cols(t): 256 b128 (stride 16 is conflict-free)
  async_b128(shk + (tid >> 2) * 16 + (tid & 3) * 4,
             K + (size_t)(tid >> 2) * T_ + t0 + (tid & 3) * 4);
  // V tile: 16 rows(t) x 64 cols(d): 256 b128 (padded stride 80)
  async_b128(shv + (tid >> 4) * SB_STRIDE + (tid & 15) * 4,
             V + (size_t)(t0 + (tid >> 4)) * D_ + (tid & 15) * 4);
}

__global__ void __launch_bounds__(256) attn_kernel(const float* __restrict__ q_ws,
                                                   const float* __restrict__ kfull,
                                                   const float* __restrict__ vfull,
                                                   float* __restrict__ ctx_ws) {
  __shared__ float sh_k[2 * 64 * 16];
  __shared__ float sh_v[2 * 16 * SB_STRIDE];
  __shared__ float sh_t[8][16 * SA_STRIDE];
  const int tid = threadIdx.x;
  const int wave = tid >> 5, lane = tid & 31;
  const int l16 = lane & 15, lhalf = lane >> 4;
  float* sc_lds = sh_t[wave];

  const int bh = blockIdx.x >> 3;                 // 0..31
  const int s0 = (blockIdx.x & 7) * 128 + wave * 16;

  const float* q = q_ws + (size_t)bh * S_ * D_;   // (S,D)
  const float* K = kfull + (size_t)bh * D_ * T_;  // (D,T)
  const float* V = vfull + (size_t)bh * T_ * D_;  // (T,D)

  // Preload Q fragments for this wave's 16 rows (entire D=64), reused 96x.
  v2f qf[16];
#pragma unroll
  for (int k = 0; k < D_; k += 4) {
    const float* p = q + (size_t)(s0 + l16) * D_ + k + 2 * lhalf;
    qf[k >> 2].x = p[0];
    qf[k >> 2].y = p[1];
  }

  v8f zero = {};
  v8f ctxc[4], ctxn[4];
#pragma unroll
  for (int n = 0; n < 4; ++n) { ctxc[n] = zero; ctxn[n] = zero; }
  float rsum[8];
#pragma unroll
  for (int r = 0; r < 8; ++r) rsum[r] = 0.f;

  issue_attn_stage(K, V, 0, sh_k, sh_v);

  const int NT = T_ / 16;                          // 96 tiles
  for (int ti = 0; ti < NT; ++ti) {
    const int t0 = ti * 16;
    const int buf = ti & 1;
    wait_async0();
    __syncthreads();
    if (ti + 1 < NT)
      issue_attn_stage(K, V, t0 + 16,
                       sh_k + (buf ^ 1) * (64 * 16),
                       sh_v + (buf ^ 1) * (16 * SB_STRIDE));
    const float* sk = sh_k + buf * (64 * 16);
    const float* sv = sh_v + buf * (16 * SB_STRIDE);

    // --- scores tile (16x16) over K-dim D=64, B operand from LDS
    v8f sc = zero;
#pragma unroll
    for (int k = 0; k < D_; k += 4) {
      v2f b;
      const float* bp = sk + (k + 2 * lhalf) * 16 + l16;
      b.x = bp[0];
      b.y = bp[16];
      sc = wmma4(qf[k >> 2], b, sc);
    }

    const bool isNew = (t0 >= CACHE_);             // wave-uniform
    if (isNew) {
      const int c_abs = (t0 - CACHE_) + l16;
#pragma unroll
      for (int r = 0; r < 8; ++r) {
        const int row_abs = s0 + r + 8 * lhalf;
        const float e = (c_abs <= row_abs) ? __expf(sc[r] * 0.125f) : 0.f;
        rsum[r] += e;
        sc_lds[(r + 8 * lhalf) * SA_STRIDE + l16] = e;
      }
    } else {
#pragma unroll
      for (int r = 0; r < 8; ++r)
        sc_lds[(r + 8 * lhalf) * SA_STRIDE + l16] = sc[r];
    }
    asm volatile("s_wait_dscnt 0x0" ::: "memory"); // LDS RAW within wave

    // --- ctx += att(16x16) @ V[t0:t0+16, 0:64] (B operand from LDS)
    if (isNew) {
#pragma unroll
      for (int kk = 0; kk < 16; kk += 4) {
        v2f a;
        a.x = sc_lds[l16 * SA_STRIDE + kk + 2 * lhalf];
        a.y = sc_lds[l16 * SA_STRIDE + kk + 2 * lhalf + 1];
#pragma unroll
        for (int n = 0; n < 4; ++n) {
          v2f b;
          const float* bp = sv + (kk + 2 * lhalf) * SB_STRIDE + 16 * n + l16;
          b.x = bp[0];
          b.y = bp[SB_STRIDE];
          ctxn[n] = wmma4(a, b, ctxn[n]);
        }
      }
    } else {
#pragma unroll
      for (int kk = 0; kk < 16; kk += 4) {
        v2f a;
        a.x = sc_lds[l16 * SA_STRIDE + kk + 2 * lhalf];
        a.y = sc_lds[l16 * SA_STRIDE + kk + 2 * lhalf + 1];
#pragma unroll
        for (int n = 0; n < 4; ++n) {
          v2f b;
          const float* bp = sv + (kk + 2 * lhalf) * SB_STRIDE + 16 * n + l16;
          b.x = bp[0];
          b.y = bp[SB_STRIDE];
          ctxc[n] = wmma4(a, b, ctxc[n]);
        }
      }
    }
  }

  // --- row-sum reduction across 16 lanes holding each row
#pragma unroll
  for (int r = 0; r < 8; ++r) {
    float s = rsum[r];
#pragma unroll
    for (int off = 1; off < 16; off <<= 1) s += __shfl_xor(s, off, 32);
    rsum[r] = 1.0f / (s + 1e-5f);
  }

  // --- store ctx permuted: (B,H,S,D) -> (B,S,D,H) flat (B,S,E)
  const int b = bh / H_, h = bh % H_;
#pragma unroll
  for (int n = 0; n < 4; ++n) {
    const int d = 16 * n + l16;
#pragma unroll
    for (int r = 0; r < 8; ++r) {
      const int srow = s0 + r + 8 * lhalf;
      const float val = ctxc[n][r] + ctxn[n][r] * rsum[r];
      ctx_ws[((size_t)(b * S_ + srow) * D_ + d) * H_ + h] = val;
    }
  }
}

// ---------------------------------------------------------------------------
// Kernel 4: out = ctx_flat @ W_out + b_out   (2048 x 1024 x 1024)
//   Grid: 16 * 16 = 256 blocks.
// ---------------------------------------------------------------------------
__global__ void __launch_bounds__(256) outproj_kernel(const float* __restrict__ ctx,
                                                      const float* __restrict__ Wout,
                                                      const float* __restrict__ bout,
                                                      float* __restrict__ out) {
  __shared__ float sh_a[2 * 128 * SA_STRIDE];
  __shared__ float sh_b[2 * 16 * SB_STRIDE];
  const int wave = threadIdx.x >> 5;
  const int lane = threadIdx.x & 31;
  const int l16 = lane & 15, lhalf = lane >> 4;
  const int nCB = E_ / 64;                 // 16
  const int row0 = (blockIdx.x / nCB) * 128;
  const int col0 = (blockIdx.x % nCB) * 64;

  v8f zero = {};
  v8f acc[4];
#pragma unroll
  for (int n = 0; n < 4; ++n) acc[n] = zero;

  gemm_block_128x64(ctx, E_, Wout, E_, E_, row0, col0, sh_a, sh_b,
                    wave, l16, lhalf, acc);

  const int rowW = row0 + wave * 16;
#pragma unroll
  for (int n = 0; n < 4; ++n) {
    const int col = col0 + 16 * n + l16;
    const float bias = bout[col];
#pragma unroll
    for (int r = 0; r < 8; ++r) {
      const int row = rowW + r + 8 * lhalf;
      out[(size_t)row * E_ + col] = acc[n][r] + bias;
    }
  }
}

// ---------------------------------------------------------------------------
extern "C" void kernel_launch(void* const* d_in, const int* in_sizes, int n_in,
                              void* d_out, int out_size, void* d_ws, size_t ws_size,
                              hipStream_t stream) {
  const float* x    = (const float*)d_in[0];
  const float* Wqkv = (const float*)d_in[1];
  const float* bqkv = (const float*)d_in[2];
  const float* Wout = (const float*)d_in[3];
  const float* bout = (const float*)d_in[4];
  const float* ck   = (const float*)d_in[5];
  const float* cv   = (const float*)d_in[6];

  float* out   = (float*)d_out;                          // (B,S,E)
  float* kfull = out + (size_t)B_ * S_ * E_;             // (B,H,D,T)
  float* vfull = kfull + (size_t)B_ * H_ * D_ * T_;      // (B,H,T,D)

  float* q_ws   = (float*)d_ws;                          // (B,H,S,D) = 8 MB
  float* ctx_ws = q_ws + (size_t)B_ * H_ * S_ * D_;      // (B,S,E)   = 8 MB

  // 1) qkv projection + scatter: 16 row-blocks x 48 col-blocks
  qkv_kernel<<<768, 256, 0, stream>>>(x, Wqkv, bqkv, q_ws, kfull, vfull);

  // 2) cache concat copies (float4)
  {
    const int total4 = (B_ * H_ * D_ * CACHE_ + B_ * H_ * CACHE_ * D_) / 4;
    copy_cache_kernel<<<(total4 + 255) / 256, 256, 0, stream>>>(
        (const float4*)ck, (const float4*)cv, kfull, vfull);
  }

  // 3) attention: 32 (b,h) x 8 s-blocks = 256 blocks
  attn_kernel<<<256, 256, 0, stream>>>(q_ws, kfull, vfull, ctx_ws);

  // 4) output projection: 16 x 16 = 256 blocks
  outproj_kernel<<<256, 256, 0, stream>>>(ctx_ws, Wout, bout, out);
}